// MultiHeadAttention_wpos_41137196761094
// MI455X (gfx1250) — compile-verified
//
#include <hip/hip_runtime.h>
#include <math.h>

#define SEQ 2048
#define DM 768
#define NH 12
#define DEPTH 64
#define FEAT 128
#define PPOS (2*SEQ-1)   // 4095

typedef float v2f __attribute__((ext_vector_type(2)));
typedef float v4f __attribute__((ext_vector_type(4)));
typedef float v8f __attribute__((ext_vector_type(8)));

// ---------------------------------------------------------------------------
// Positional features (Enformer basis): pe[j, 0:768], j in [0, 4095)
// ---------------------------------------------------------------------------
__global__ void pos_feat_kernel(float* __restrict__ pe) {
    int idx = blockIdx.x * blockDim.x + threadIdx.x;
    if (idx >= PPOS * FEAT) return;
    int j = idx / FEAT;
    int f = idx % FEAT;
    float pos  = (float)(j - (SEQ - 1));
    float apos = fabsf(pos);
    float sgn  = (pos > 0.f) ? 1.f : ((pos < 0.f) ? -1.f : 0.f);

    // exponential basis: half_life = 2^linspace(3, log2(2048)=11, 128)
    float e  = 3.0f + 8.0f * (float)f / 127.0f;
    float hl = exp2f(e);
    float ex = expf(-0.6931471805599453f / hl * apos);

    // central mask basis
    float cw = exp2f((float)(f + 1)) - 1.0f;
    float cm = (cw > apos) ? 1.0f : 0.0f;

    // gamma pdf basis (unnormalized; global max applied later)
    float mean = 16.0f + (2048.0f - 16.0f) * (float)f / 127.0f;  // linspace(16,2048,128)
    float conc = (mean * 0.125f) * (mean * 0.125f);              // (mean/8)^2
    float rate = mean * (1.0f / 64.0f);                          // mean/64
    float prob = 0.0f;
    if (apos > 0.f) {
        float lu = (conc - 1.0f) * logf(apos) - rate * apos;
        float ln = lgammaf(conc) - conc * logf(rate);
        prob = expf(lu - ln);
    }
    size_t base = (size_t)j * DM;
    pe[base + 0*FEAT + f] = ex;
    pe[base + 1*FEAT + f] = ex * sgn;
    pe[base + 2*FEAT + f] = cm;
    pe[base + 3*FEAT + f] = cm * sgn;
    pe[base + 4*FEAT + f] = prob;
    pe[base + 5*FEAT + f] = 0.0f;
}

__global__ void init_max_kernel(float* m) { *(int*)m = 0; }

__global__ void max_reduce_kernel(const float* __restrict__ pe, float* __restrict__ maxp) {
    __shared__ float red[256];
    float m = 0.f;
    int total = PPOS * FEAT;
    for (int i = blockIdx.x * blockDim.x + threadIdx.x; i < total; i += gridDim.x * blockDim.x) {
        int j = i / FEAT, f = i % FEAT;
        m = fmaxf(m, pe[(size_t)j * DM + 4*FEAT + f]);
    }
    red[threadIdx.x] = m; __syncthreads();
    for (int s = 128; s > 0; s >>= 1) {
        if ((int)threadIdx.x < s) red[threadIdx.x] = fmaxf(red[threadIdx.x], red[threadIdx.x + s]);
        __syncthreads();
    }
    if (threadIdx.x == 0) atomicMax((int*)maxp, __float_as_int(red[0]));  // prob >= 0
}

__global__ void pos_finalize_kernel(float* __restrict__ pe, const float* __restrict__ maxp) {
    int idx = blockIdx.x * blockDim.x + threadIdx.x;
    if (idx >= PPOS * FEAT) return;
    int j = idx / FEAT, f = idx % FEAT;
    float inv = 1.0f / (*maxp);
    float pos = (float)(j - (SEQ - 1));
    float sgn = (pos > 0.f) ? 1.f : ((pos < 0.f) ? -1.f : 0.f);
    size_t base = (size_t)j * DM;
    float g = pe[base + 4*FEAT + f] * inv;
    pe[base + 4*FEAT + f] = g;
    pe[base + 5*FEAT + f] = g * sgn;
}

// ---------------------------------------------------------------------------
// fp32 WMMA GEMM:  C = alpha * A @ op(B)  (+ bias[col]) (+ shifted rel)
//   TRANS_B=false: B is [K,N] row-major (ldb); true: B is [N,K] row-major.
//   relAdd != null: C[r,c] += relAdd[r*relLd + relOff + c - r]  (rel-shift fuse)
//   blockIdx.z batches: A += z*strA, B += z*strB, C += z*strC.
// Tile 64x64, 128 threads = 4 waves, 2x2 v_wmma_f32_16x16x4_f32 per wave.
// Double-buffered LDS (M-major, stride 20 -> contiguous v2f fragments,
// conflict-free ds_load_b64) + register prefetch of the next K-tile.
// ---------------------------------------------------------------------------
template<bool TRANS_B>
__global__ __launch_bounds__(128)
void gemm_wmma(const float* __restrict__ A, const float* __restrict__ B,
               const float* __restrict__ bias,
               const float* __restrict__ relAdd, int relLd, int relOff,
               float* __restrict__ C,
               int M, int N, int K, int lda, int ldb, int ldc, float alpha,
               long strA, long strB, long strC)
{
    __shared__ float As[2][64][20];
    __shared__ float Bs[2][64][20];

    A += (size_t)blockIdx.z * strA;
    B += (size_t)blockIdx.z * strB;
    C += (size_t)blockIdx.z * strC;

    const int tid  = threadIdx.x;
    const int lane = tid & 31;
    const int wave = tid >> 5;
    const int half = lane >> 4;    // 0: lanes 0-15, 1: lanes 16-31
    const int l16  = lane & 15;
    const int wm = (wave >> 1) * 32;
    const int wn = (wave & 1) * 32;
    const int row0 = blockIdx.y * 64;
    const int col0 = blockIdx.x * 64;

    // A (and NT-B) prefetch: 256 vec4 slots; this thread owns rows am, am+32, cols [ak, ak+4)
    const int am = tid >> 2;
    const int ak = (tid & 3) * 4;
    // NN-B prefetch: 8 scalars; k = bk + 2j, n = bn
    const int bn  = tid & 63;
    const int bk  = tid >> 6;

    v4f pa0, pa1, pb0, pb1;
    float rbs[8];
    const v4f vz = {0.f, 0.f, 0.f, 0.f};

    auto loadTile = [&](int k0) {
        int g0 = row0 + am, g1 = row0 + am + 32;
        pa0 = (g0 < M) ? *(const v4f*)&A[(size_t)g0 * lda + k0 + ak] : vz;
        pa1 = (g1 < M) ? *(const v4f*)&A[(size_t)g1 * lda + k0 + ak] : vz;
        if (TRANS_B) {
            int c0 = col0 + am, c1 = col0 + am + 32;
            pb0 = (c0 < N) ? *(const v4f*)&B[(size_t)c0 * ldb + k0 + ak] : vz;
            pb1 = (c1 < N) ? *(const v4f*)&B[(size_t)c1 * ldb + k0 + ak] : vz;
        } else {
            int gn = col0 + bn;
            bool ok = (gn < N);
#pragma unroll
            for (int j = 0; j < 8; ++j)
                rbs[j] = ok ? B[(size_t)(k0 + bk + 2*j) * ldb + gn] : 0.f;
        }
    };
    auto storeTile = [&](int buf) {
        *(v4f*)&As[buf][am][ak]      = pa0;
        *(v4f*)&As[buf][am + 32][ak] = pa1;
        if (TRANS_B) {
            *(v4f*)&Bs[buf][am][ak]      = pb0;
            *(v4f*)&Bs[buf][am + 32][ak] = pb1;
        } else {
#pragma unroll
            for (int j = 0; j < 8; ++j)
                Bs[buf][bn][bk + 2*j] = rbs[j];
        }
    };

    v8f acc[2][2] = {};
    const int ktiles = K >> 4;   // K is always a multiple of 16 here

    loadTile(0);
    storeTile(0);
    __syncthreads();

    for (int t = 0; t < ktiles; ++t) {
        const int cur = t & 1;
        if (t + 1 < ktiles) loadTile((t + 1) << 4);   // global_load_b128 in flight

        v2f af[4][2], bf[4][2];
#pragma unroll
        for (int kk = 0; kk < 4; ++kk) {
            const int ka = kk * 4 + half * 2;         // A: reg r -> K = half*2 + r
            af[kk][0] = *(const v2f*)&As[cur][wm + l16][ka];
            af[kk][1] = *(const v2f*)&As[cur][wm + 16 + l16][ka];
            bf[kk][0] = *(const v2f*)&Bs[cur][wn + l16][ka];
            bf[kk][1] = *(const v2f*)&Bs[cur][wn + 16 + l16][ka];
        }
#pragma unroll
        for (int kk = 0; kk < 4; ++kk) {
            acc[0][0] = __builtin_amdgcn_wmma_f32_16x16x4_f32(false, af[kk][0], false, bf[kk][0], (short)0, acc[0][0], false, false);
            acc[0][1] = __builtin_amdgcn_wmma_f32_16x16x4_f32(false, af[kk][0], false, bf[kk][1], (short)0, acc[0][1], false, false);
            acc[1][0] = __builtin_amdgcn_wmma_f32_16x16x4_f32(false, af[kk][1], false, bf[kk][0], (short)0, acc[1][0], false, false);
            acc[1][1] = __builtin_amdgcn_wmma_f32_16x16x4_f32(false, af[kk][1], false, bf[kk][1], (short)0, acc[1][1], false, false);
        }
        if (t + 1 < ktiles) storeTile((t + 1) & 1);
        __syncthreads();
    }

    // C/D 16x16 layout: reg v, lanes 0-15 -> M=v, lanes 16-31 -> M=v+8; N=lane%16
#pragma unroll
    for (int mt = 0; mt < 2; ++mt)
#pragma unroll
        for (int nt = 0; nt < 2; ++nt)
#pragma unroll
            for (int vv = 0; vv < 8; ++vv) {
                int r = row0 + wm + mt * 16 + half * 8 + vv;
                int c = col0 + wn + nt * 16 + l16;
                if (r < M && c < N) {
                    float val = alpha * acc[mt][nt][vv];
                    if (bias)   val += bias[c];
                    if (relAdd) val += relAdd[(size_t)r * relLd + (relOff + c - r)];
                    C[(size_t)r * ldc + c] = val;
                }
            }
}

// ---------------------------------------------------------------------------
// Row softmax in place, b128 streaming: one block per row
// ---------------------------------------------------------------------------
__global__ void softmax_rows(float* __restrict__ a, int ncols) {
    __shared__ float red[256];
    v4f* row = (v4f*)(a + (size_t)blockIdx.x * ncols);
    const int n4 = ncols >> 2;
    const int t = threadIdx.x;

    float m = -INFINITY;
    for (int i = t; i < n4; i += 256) {
        v4f x = row[i];
        m = fmaxf(m, fmaxf(fmaxf(x.x, x.y), fmaxf(x.z, x.w)));
    }
    red[t] = m; __syncthreads();
    for (int s = 128; s > 0; s >>= 1) { if (t < s) red[t] = fmaxf(red[t], red[t + s]); __syncthreads(); }
    m = red[0]; __syncthreads();

    float sum = 0.f;
    for (int i = t; i < n4; i += 256) {
        v4f x = row[i];
        x.x = expf(x.x - m); x.y = expf(x.y - m);
        x.z = expf(x.z - m); x.w = expf(x.w - m);
        row[i] = x;
        sum += (x.x + x.y) + (x.z + x.w);
    }
    red[t] = sum; __syncthreads();
    for (int s = 128; s > 0; s >>= 1) { if (t < s) red[t] += red[t + s]; __syncthreads(); }
    const float inv = 1.0f / red[0];

    for (int i = t; i < n4; i += 256) {
        v4f x = row[i];
        x.x *= inv; x.y *= inv; x.z *= inv; x.w *= inv;
        row[i] = x;
    }
}

// ---------------------------------------------------------------------------
extern "C" void kernel_launch(void* const* d_in, const int* in_sizes, int n_in,
                              void* d_out, int out_size, void* d_ws, size_t ws_size,
                              hipStream_t stream)
{
    const float* q     = (const float*)d_in[0];
    const float* kin   = (const float*)d_in[1];
    const float* vin   = (const float*)d_in[2];
    const float* wq    = (const float*)d_in[3];
    const float* wk    = (const float*)d_in[4];
    const float* wv    = (const float*)d_in[5];
    const float* w_rk  = (const float*)d_in[6];
    const float* r_w   = (const float*)d_in[7];   // [H*DEPTH] per-column bias
    const float* r_r   = (const float*)d_in[8];
    const float* w_out = (const float*)d_in[9];
    const float* b_out = (const float*)d_in[10];

    float* out  = (float*)d_out;                     // [SEQ, DM]
    float* attn = out + (size_t)SEQ * DM;            // [NH, SEQ, SEQ]

    float* ws   = (float*)d_ws;
    float* pe   = ws;                                // PPOS*DM
    float* rk   = pe  + (size_t)PPOS * DM;           // PPOS*DM
    float* qw   = rk  + (size_t)PPOS * DM;           // SEQ*DM   (qh/8 + r_w)
    float* qr   = qw  + (size_t)SEQ * DM;            // SEQ*DM   (qh/8 + r_r)
    float* kh   = qr  + (size_t)SEQ * DM;            // SEQ*DM
    float* vh   = kh  + (size_t)SEQ * DM;            // SEQ*DM
    float* ov   = vh  + (size_t)SEQ * DM;            // SEQ*DM   (heads-concat attn@V)
    float* rel  = ov  + (size_t)SEQ * DM;            // SEQ*PPOS (per-head, reused)
    float* maxp = rel + (size_t)SEQ * PPOS;          // 1

    // 1) positional features (incl. global-max normalization of gamma basis)
    int pf_total = PPOS * FEAT;
    pos_feat_kernel<<<(pf_total + 255) / 256, 256, 0, stream>>>(pe);
    init_max_kernel<<<1, 1, 0, stream>>>(maxp);
    max_reduce_kernel<<<512, 256, 0, stream>>>(pe, maxp);
    pos_finalize_kernel<<<(pf_total + 255) / 256, 256, 0, stream>>>(pe, maxp);

    dim3 blk(128);
    // 2) rk = pe @ w_rk                       [PPOS, DM]
    gemm_wmma<false><<<dim3(DM/64, (PPOS+63)/64, 1), blk, 0, stream>>>(
        pe, w_rk, nullptr, nullptr, 0, 0, rk, PPOS, DM, DM, DM, DM, DM, 1.0f, 0, 0, 0);
    // 3) projections: qw = (q@wq)/8 + r_w ; qr = (q@wq)/8 + r_r ; kh, vh
    gemm_wmma<false><<<dim3(DM/64, SEQ/64, 1), blk, 0, stream>>>(
        q, wq, r_w, nullptr, 0, 0, qw, SEQ, DM, DM, DM, DM, DM, 0.125f, 0, 0, 0);
    gemm_wmma<false><<<dim3(DM/64, SEQ/64, 1), blk, 0, stream>>>(
        q, wq, r_r, nullptr, 0, 0, qr, SEQ, DM, DM, DM, DM, DM, 0.125f, 0, 0, 0);
    gemm_wmma<false><<<dim3(DM/64, SEQ/64, 1), blk, 0, stream>>>(
        kin, wk, nullptr, nullptr, 0, 0, kh, SEQ, DM, DM, DM, DM, DM, 1.0f, 0, 0, 0);
    gemm_wmma<false><<<dim3(DM/64, SEQ/64, 1), blk, 0, stream>>>(
        vin, wv, nullptr, nullptr, 0, 0, vh, SEQ, DM, DM, DM, DM, DM, 1.0f, 0, 0, 0);

    // 4) per-head scores: rel logits, then content NT-GEMM with fused rel-shift add
    //    (rel ws buffer reused per head; stream order serializes correctly)
    for (int h = 0; h < NH; ++h) {
        float* sc = attn + (size_t)h * SEQ * SEQ;
        // rel[s,j] = qr_h[s,:] . rk_h[j,:]     [SEQ, PPOS]
        gemm_wmma<true><<<dim3((PPOS+63)/64, SEQ/64, 1), blk, 0, stream>>>(
            qr + h*DEPTH, rk + h*DEPTH, nullptr, nullptr, 0, 0,
            rel, SEQ, PPOS, DEPTH, DM, DM, PPOS, 1.0f, 0, 0, 0);
        // sc[s,t] = qw_h[s,:] . kh_h[t,:] + rel[s, SEQ-1 + t - s]
        gemm_wmma<true><<<dim3(SEQ/64, SEQ/64, 1), blk, 0, stream>>>(
            qw + h*DEPTH, kh + h*DEPTH, nullptr, rel, PPOS, SEQ-1,
            sc, SEQ, SEQ, DEPTH, DM, DM, SEQ, 1.0f, 0, 0, 0);
    }

    // 5) softmax in place over attn rows
    softmax_rows<<<NH * SEQ, 256, 0, stream>>>(attn, SEQ);

    // 6) attn @ V for all heads in one batched launch (z = head)
    gemm_wmma<false><<<dim3(1, SEQ/64, NH), blk, 0, stream>>>(
        attn, vh, nullptr, nullptr, 0, 0,
        ov, SEQ, DEPTH, SEQ, SEQ, DM, DM, 1.0f,
        (long)SEQ * SEQ, (long)DEPTH, (long)DEPTH);

    // 7) out = ov @ w_out + b_out
    gemm_wmma<false><<<dim3(DM/64, SEQ/64, 1), blk, 0, stream>>>(
        ov, w_out, b_out, nullptr, 0, 0, out, SEQ, DM, DM, DM, DM, DM, 1.0f, 0, 0, 0);
}